// MaskedIoU_34565896798610
// MI455X (gfx1250) — compile-verified
//
#include <hip/hip_runtime.h>

// MaskedIoU for output[16,21,512,512] f32, target[16,512,512] int32.
// Memory-bound: 352MB single-pass stream -> NT loads; wave32 ballot
// histograms; WMMA (f32 16x16x4) used for the final class-sum reduction.

#define NCLS 21
#define HW_SHIFT 18
#define HW (1 << HW_SHIFT)        // 512*512
#define CHW (NCLS * HW)
#define WS_PAD 32                 // 128B between global counters (atomic spread)

typedef __attribute__((ext_vector_type(4))) float f4;
typedef __attribute__((ext_vector_type(4))) int   i4;
typedef __attribute__((ext_vector_type(2))) float v2f;
typedef __attribute__((ext_vector_type(8))) float v8f;

static __device__ __forceinline__ unsigned bal32(bool p) {
#if __has_builtin(__builtin_amdgcn_ballot_w32)
  return __builtin_amdgcn_ballot_w32(p);
#else
  return (unsigned)__ballot(p);
#endif
}

__global__ void iou_init_kernel(unsigned int* __restrict__ ws, int wsStride) {
  int i = threadIdx.x;
  if (i < 3 * NCLS) ws[i * wsStride] = 0u;
}

__global__ __launch_bounds__(256) void iou_hist_kernel(
    const float* __restrict__ logits, const int* __restrict__ target,
    unsigned int* __restrict__ ws, int wsStride, int n4) {
  __shared__ unsigned int h[3 * NCLS];
  for (int i = threadIdx.x; i < 3 * NCLS; i += blockDim.x) h[i] = 0u;
  __syncthreads();

  const unsigned lane = threadIdx.x & 31u;
  unsigned accTp = 0u, accPr = 0u, accTg = 0u;

  const int nthreads = (int)(gridDim.x * blockDim.x);
  const int iters = (n4 + nthreads - 1) / nthreads;
  int g = (int)(blockIdx.x * blockDim.x + threadIdx.x);

  for (int it = 0; it < iters; ++it, g += nthreads) {
    int bi0 = -1, bi1 = -1, bi2 = -1, bi3 = -1;
    int t0 = -2, t1 = -2, t2 = -2, t3 = -2;
    bool ok0 = false, ok1 = false, ok2 = false, ok3 = false;

    if (g < n4) {
      const int p  = g << 2;                 // first pixel of this 4-group
      const int b  = p >> HW_SHIFT;          // batch index
      const int hw = p & (HW - 1);
      const float* __restrict__ base = logits + (size_t)b * CHW + hw;

      // argmax over 21 classes, 4 pixels at a time (B128 NT loads)
      f4 best = __builtin_nontemporal_load((const f4*)base);
      bi0 = bi1 = bi2 = bi3 = 0;
#pragma unroll
      for (int c = 1; c < NCLS; ++c) {
        f4 v = __builtin_nontemporal_load((const f4*)(base + (size_t)c * HW));
        if (v.x > best.x) { best.x = v.x; bi0 = c; }
        if (v.y > best.y) { best.y = v.y; bi1 = c; }
        if (v.z > best.z) { best.z = v.z; bi2 = c; }
        if (v.w > best.w) { best.w = v.w; bi3 = c; }
      }
      i4 t = __builtin_nontemporal_load((const i4*)target + g);
      t0 = t.x; t1 = t.y; t2 = t.z; t3 = t.w;
      ok0 = (bi0 == t0); ok1 = (bi1 == t1); ok2 = (bi2 == t2); ok3 = (bi3 == t3);
    }

    // Wave-wide ballot histograms; lane c owns the counters for class c.
    // All lanes reach here every iteration (invalid lanes contribute 0 bits).
#pragma unroll
    for (int c = 0; c < NCLS; ++c) {
      unsigned pr = __popc(bal32(bi0 == c)) + __popc(bal32(bi1 == c)) +
                    __popc(bal32(bi2 == c)) + __popc(bal32(bi3 == c));
      unsigned tg = __popc(bal32(t0 == c)) + __popc(bal32(t1 == c)) +
                    __popc(bal32(t2 == c)) + __popc(bal32(t3 == c));
      unsigned tp = __popc(bal32(ok0 && (t0 == c))) + __popc(bal32(ok1 && (t1 == c))) +
                    __popc(bal32(ok2 && (t2 == c))) + __popc(bal32(ok3 && (t3 == c)));
      if (lane == (unsigned)c) { accPr += pr; accTg += tg; accTp += tp; }
    }
  }

  if (lane < NCLS) {
    atomicAdd(&h[lane],            accTp);   // [0..20]  tp
    atomicAdd(&h[NCLS + lane],     accPr);   // [21..41] pred count
    atomicAdd(&h[2 * NCLS + lane], accTg);   // [42..62] target count
  }
  __syncthreads();
  for (int i = threadIdx.x; i < 3 * NCLS; i += blockDim.x)
    atomicAdd(&ws[i * wsStride], h[i]);
}

// One wave (32 threads, EXEC all-ones). Per-class IoU, then the 21-element
// sum via v_wmma_f32_16x16x4_f32: with A==all-ones, D[m][n] = sum_k B[k][n],
// so reducing D's VGPR0 across the wave yields 2 * sum(all elements of B)
// independent of the exact A/B register striping.
__global__ __launch_bounds__(32) void iou_finalize_kernel(
    const unsigned int* __restrict__ ws, int wsStride, float* __restrict__ out) {
  const unsigned c = threadIdx.x;                     // 0..31
  const int cc = (c < NCLS) ? (int)c : 0;             // clamp: in-bounds reads
  const float tp = (float)ws[cc * wsStride];
  const float pr = (float)ws[(NCLS + cc) * wsStride];
  const float tg = (float)ws[(2 * NCLS + cc) * wsStride];
  const float uni = pr + tg - tp;
  const float iou = (c < NCLS && uni > 0.0f) ? (tp / uni) : 0.0f;

  v2f a; a.x = 1.0f; a.y = 1.0f;                      // A = all ones (16x4)
  v2f b; b.x = iou;  b.y = 0.0f;                      // B holds the 21 IoUs
  v8f acc = {};
  acc = __builtin_amdgcn_wmma_f32_16x16x4_f32(
      /*neg_a=*/false, a, /*neg_b=*/false, b,
      /*c_mod=*/(short)0, acc, /*reuse_a=*/false, /*reuse_b=*/false);

  float s = acc[0];                                   // D VGPR0
#pragma unroll
  for (int off = 16; off >= 1; off >>= 1) s += __shfl_xor(s, off, 32);
  if (c == 0) out[0] = s * (0.5f / (float)NCLS);      // wave sum = 2*Sigma(iou)
}

extern "C" void kernel_launch(void* const* d_in, const int* in_sizes, int n_in,
                              void* d_out, int out_size, void* d_ws, size_t ws_size,
                              hipStream_t stream) {
  const float* logits = (const float*)d_in[0];
  const int*   target = (const int*)d_in[1];
  float*       res    = (float*)d_out;
  unsigned int* ws    = (unsigned int*)d_ws;

  const int N  = in_sizes[1];       // 16*512*512 pixels
  const int n4 = N >> 2;            // 4-pixel groups

  // Pad counters 128B apart if the workspace allows (spreads L2 atomic traffic).
  const int wsStride =
      (ws_size >= (size_t)(3 * NCLS * WS_PAD * sizeof(unsigned int))) ? WS_PAD : 1;

  iou_init_kernel<<<1, 64, 0, stream>>>(ws, wsStride);
  iou_hist_kernel<<<512, 256, 0, stream>>>(logits, target, ws, wsStride, n4);
  iou_finalize_kernel<<<1, 32, 0, stream>>>(ws, wsStride, res);
}